// CosinePairwiseLoss_6768868458994
// MI455X (gfx1250) — compile-verified
//
#include <hip/hip_runtime.h>
#include <hip/hip_bf16.h>

// Problem constants (match reference).
#define N_ROWS 16384
#define DIMS   64
#define NCLS   64

typedef __attribute__((ext_vector_type(2))) float v2f;
typedef __attribute__((ext_vector_type(8))) float v8f;

#if defined(__has_builtin)
#if __has_builtin(__builtin_amdgcn_wmma_f32_16x16x4_f32)
#define USE_WMMA_F32 1
#endif
#endif

// ---------------------------------------------------------------------------
// Kernel 0: zero the accumulation regions of the workspace (S and counts).
// d_ws is poisoned 0xAA by the harness and never re-cleared between replays,
// so every kernel_launch must re-zero its accumulators.
// ---------------------------------------------------------------------------
__global__ void cpl_init_zero(float* __restrict__ S,
                              unsigned int* __restrict__ counts) {
  for (int i = threadIdx.x; i < NCLS * DIMS; i += blockDim.x) S[i] = 0.0f;
  if (threadIdx.x < NCLS) counts[threadIdx.x] = 0u;
}

// ---------------------------------------------------------------------------
// Kernel 1: row-normalize. One wave32 per row (D=64 -> float2 per lane).
// Also accumulates per-class sample counts.
// ---------------------------------------------------------------------------
__global__ void __launch_bounds__(256)
cpl_normalize(const float* __restrict__ feat,
              const int* __restrict__ pred,
              float* __restrict__ fnorm,
              unsigned int* __restrict__ counts) {
  const int row  = (int)((blockIdx.x * blockDim.x + threadIdx.x) >> 5);
  const int lane = (int)(threadIdx.x & 31);
  if (row >= N_ROWS) return;

  float2 v = ((const float2*)(feat + (size_t)row * DIMS))[lane];
  float ss = v.x * v.x + v.y * v.y;
  // wave32 butterfly reduction
  #pragma unroll
  for (int off = 16; off > 0; off >>= 1)
    ss += __shfl_xor(ss, off, 32);
  const float inv = rsqrtf(ss);
  float2 o; o.x = v.x * inv; o.y = v.y * inv;
  ((float2*)(fnorm + (size_t)row * DIMS))[lane] = o;

  if (lane == 0) atomicAdd(&counts[pred[row]], 1u);
}

// ---------------------------------------------------------------------------
// Kernel 2: S[c][d] = sum over rows of one-hot(pred) (x) fnorm, as a GEMM
//   S (64x64) = M^T (64xN) * F (Nx64),  M = one-hot class matrix (exact f32).
// Mapped on V_WMMA_F32_16X16X4_F32: per wave, one 16-dim block, four 16-class
// tiles, K strip-mined 4 rows at a time.
//
// f32 A (16x4) layout : lane L -> M = L&15, VGPR0 = K=(L>>4)*2, VGPR1 = K+1
// f32 B (4x16) layout : lane L -> N = L&15, VGPR0 = K=(L>>4)*2, VGPR1 = K+1
// f32 C/D (16x16)     : VGPR v, lane L -> M = v + (L>>4)*8, N = L&15
// ---------------------------------------------------------------------------
__global__ void __launch_bounds__(256)
cpl_segsum(const float* __restrict__ fnorm,
           const int* __restrict__ pred,
           float* __restrict__ S) {
  __shared__ float Sl[NCLS * DIMS];
  for (int i = threadIdx.x; i < NCLS * DIMS; i += blockDim.x) Sl[i] = 0.0f;
  __syncthreads();

  const int wave = (int)(threadIdx.x >> 5); // 0..7
  const int lane = (int)(threadIdx.x & 31);
  const int db   = wave & 3;                // dim block 0..3
  const int kg   = wave >> 2;               // K-stream group 0..1
  const int d0   = db * 16;
  const int n    = lane & 15;               // N for B/C/D, M for A
  const int kb   = (lane >> 4) * 2;         // K sub-offset 0 or 2

  v8f acc0 = {}, acc1 = {}, acc2 = {}, acc3 = {};

  const int nchunks = N_ROWS / 4;
  const int first   = (int)blockIdx.x * 2 + kg;
  const int stride  = (int)gridDim.x * 2;

  for (int chunk = first; chunk < nchunks; chunk += stride) {
    const int i0 = chunk * 4;
    const int r0 = i0 + kb;
    const int p0 = pred[r0];
    const int p1 = pred[r0 + 1];
    v2f b;
    b.x = fnorm[(size_t)r0 * DIMS + d0 + n];
    b.y = fnorm[(size_t)(r0 + 1) * DIMS + d0 + n];
#if defined(USE_WMMA_F32)
    {
      v2f a; a.x = (p0 == (n))      ? 1.0f : 0.0f; a.y = (p1 == (n))      ? 1.0f : 0.0f;
      acc0 = __builtin_amdgcn_wmma_f32_16x16x4_f32(false, a, false, b, (short)0, acc0, false, false);
    }
    {
      v2f a; a.x = (p0 == (16 + n)) ? 1.0f : 0.0f; a.y = (p1 == (16 + n)) ? 1.0f : 0.0f;
      acc1 = __builtin_amdgcn_wmma_f32_16x16x4_f32(false, a, false, b, (short)0, acc1, false, false);
    }
    {
      v2f a; a.x = (p0 == (32 + n)) ? 1.0f : 0.0f; a.y = (p1 == (32 + n)) ? 1.0f : 0.0f;
      acc2 = __builtin_amdgcn_wmma_f32_16x16x4_f32(false, a, false, b, (short)0, acc2, false, false);
    }
    {
      v2f a; a.x = (p0 == (48 + n)) ? 1.0f : 0.0f; a.y = (p1 == (48 + n)) ? 1.0f : 0.0f;
      acc3 = __builtin_amdgcn_wmma_f32_16x16x4_f32(false, a, false, b, (short)0, acc3, false, false);
    }
#else
    // Exact fallback: every lane owns a distinct (row, dim) pair.
    atomicAdd(&Sl[p0 * DIMS + d0 + n], b.x);
    atomicAdd(&Sl[p1 * DIMS + d0 + n], b.y);
#endif
  }

#if defined(USE_WMMA_F32)
  const int mhi = (lane >> 4) * 8;
  #pragma unroll
  for (int v = 0; v < 8; ++v) {
    atomicAdd(&Sl[(0  + v + mhi) * DIMS + d0 + n], acc0[v]);
    atomicAdd(&Sl[(16 + v + mhi) * DIMS + d0 + n], acc1[v]);
    atomicAdd(&Sl[(32 + v + mhi) * DIMS + d0 + n], acc2[v]);
    atomicAdd(&Sl[(48 + v + mhi) * DIMS + d0 + n], acc3[v]);
  }
#endif
  __syncthreads();
  for (int i = threadIdx.x; i < NCLS * DIMS; i += blockDim.x)
    atomicAdd(&S[i], Sl[i]);
}

// ---------------------------------------------------------------------------
// Kernel 3: finalize. One thread per class:
//   pair_sum_c = 0.5*(||S_c||^2 - n_c), pair_cnt_c = n_c*(n_c-1)/2
//   out = 1 - (sum_c avg_c) / #unique
// ---------------------------------------------------------------------------
__global__ void cpl_finalize(const float* __restrict__ S,
                             const unsigned int* __restrict__ counts,
                             float* __restrict__ out) {
  __shared__ float s_avg[NCLS];
  __shared__ float s_uni[NCLS];
  const int c = (int)threadIdx.x;
  const float nc = (float)counts[c];
  float ssq = 0.0f;
  #pragma unroll 8
  for (int d = 0; d < DIMS; ++d) {
    const float x = S[c * DIMS + d];
    ssq += x * x;
  }
  const float pair_cnt = 0.5f * nc * (nc - 1.0f);
  const float pair_sum = 0.5f * (ssq - nc);
  s_avg[c] = (pair_cnt > 0.0f) ? (pair_sum / pair_cnt) : 0.0f;
  s_uni[c] = (nc > 0.0f) ? 1.0f : 0.0f;
  __syncthreads();
  if (c == 0) {
    float sa = 0.0f, su = 0.0f;
    for (int i = 0; i < NCLS; ++i) { sa += s_avg[i]; su += s_uni[i]; }
    out[0] = 1.0f - sa / su;
  }
}

// ---------------------------------------------------------------------------
// Host launcher. Workspace layout (needs ~4.21 MB):
//   [0)                fnorm  : N*D floats (4 MB)
//   [N*D)              S      : 64*64 floats
//   [N*D + 64*64)      counts : 64 uints
// ---------------------------------------------------------------------------
extern "C" void kernel_launch(void* const* d_in, const int* in_sizes, int n_in,
                              void* d_out, int out_size, void* d_ws, size_t ws_size,
                              hipStream_t stream) {
  (void)in_sizes; (void)n_in; (void)out_size; (void)ws_size;
  const float* feat = (const float*)d_in[0];
  const int*   pred = (const int*)d_in[1];
  float*       out  = (float*)d_out;

  float*        fnorm  = (float*)d_ws;
  float*        S      = fnorm + (size_t)N_ROWS * DIMS;
  unsigned int* counts = (unsigned int*)(S + NCLS * DIMS);

  cpl_init_zero<<<1, 256, 0, stream>>>(S, counts);
  cpl_normalize<<<(N_ROWS * 32) / 256, 256, 0, stream>>>(feat, pred, fnorm, counts);
  cpl_segsum<<<128, 256, 0, stream>>>(fnorm, pred, S);
  cpl_finalize<<<1, NCLS, 0, stream>>>(S, counts, out);
}